// RNNSeqDecoder_1778116460654
// MI455X (gfx1250) — compile-verified
//
#include <hip/hip_runtime.h>
#include <math.h>

typedef __attribute__((ext_vector_type(2))) float v2f;
typedef __attribute__((ext_vector_type(8))) float v8f;

#define BSZ   64
#define LSEQ  200
#define FDIM  512
#define HDIM  512
#define EDIM  512
#define ADIM  512
#define DDIM  1024   // E + F
#define G3    1536   // 3*H
#define VSZ   50257
#define TSZ   21

__device__ __forceinline__ v2f ld2(const float* p) { return *(const v2f*)p; }

// ---------------------------------------------------------------------------
// Tiled GEMM C[M,N] = A[M,K] * B[K,N] (+bias[m]), row-major, fp32 WMMA.
// LDA/LDB/LDC are compile-time so every inner-loop load is base+imm-offset.
// One wave computes MT vertically-stacked 16x16 tiles sharing the B fragment.
// M-edge rows are clamped (in-bounds loads, garbage only lands in un-stored
// output rows). Wave-uniform control flow => EXEC all 1s at every WMMA.
//
// Fragment layouts (ISA 7.12.2, f32):
//   A 16x4 : lane m=lane&15, kbase=2*(lane>>4), a[i]=A[m][k+kbase+i]
//   B 4x16 : lane n=lane&15, b[i]=B[k+kbase+i][n]
//   C 16x16: vgpr r -> M=r+8*(lane>>4), N=lane&15
// ---------------------------------------------------------------------------
template<int MT, int LDA, int LDB, int LDC, bool HAS_BIAS>
__global__ void k_gemm(const float* __restrict__ A,
                       const float* __restrict__ B,
                       float* __restrict__ C,
                       const float* __restrict__ bias,
                       int M, int N, int K)
{
  const int wave    = blockIdx.x * (blockDim.x >> 5) + (threadIdx.x >> 5);
  const int ntiles  = N >> 4;
  const int mgroups = (M + 16 * MT - 1) / (16 * MT);
  const int mtg = wave / ntiles;
  const int nt  = wave % ntiles;
  if (mtg >= mgroups) return;  // wave-uniform exit

  const int lane = threadIdx.x & 31;
  const int half = lane >> 4;       // 0 or 1
  const int mr   = lane & 15;       // A row within tile
  const int nc   = lane & 15;       // B/C col within tile
  const int kb   = half * 2;
  const int m0   = mtg * 16 * MT;
  const int n0   = nt * 16;

  const float* Arow[MT];
#pragma unroll
  for (int i = 0; i < MT; ++i) {
    int ma = m0 + i * 16 + mr;
    if (ma > M - 1) ma = M - 1;           // clamp: always in-bounds
    Arow[i] = A + (size_t)ma * LDA + kb;
  }
  const float* Bcol = B + (size_t)kb * LDB + n0 + nc;

  v8f acc[MT];
#pragma unroll
  for (int i = 0; i < MT; ++i) acc[i] = v8f{};

#pragma unroll 8
  for (int k = 0; k < K; k += 4) {
    v2f b;
    b.x = Bcol[k * LDB];
    b.y = Bcol[k * LDB + LDB];
#pragma unroll
    for (int i = 0; i < MT; ++i) {
      v2f a = ld2(Arow[i] + k);
      acc[i] = __builtin_amdgcn_wmma_f32_16x16x4_f32(
          false, a, false, b, (short)0, acc[i], false, false);
    }
  }

#pragma unroll
  for (int i = 0; i < MT; ++i) {
#pragma unroll
    for (int r = 0; r < 8; ++r) {
      int mm = m0 + i * 16 + r + half * 8;
      if (mm < M) {
        float v = acc[i][r];
        if (HAS_BIAS) v += bias[mm];
        C[(size_t)mm * LDC + n0 + nc] = v;
      }
    }
  }
}

// ---------------------------------------------------------------------------
// Attention + context + concat. One block (256 threads) per batch row b.
// ---------------------------------------------------------------------------
__global__ void k_attn(const float* __restrict__ enc_proj,
                       const float* __restrict__ hWa,
                       const float* __restrict__ v_att,
                       const float* __restrict__ mask,
                       const float* __restrict__ enc,
                       const float* __restrict__ emb,
                       const int*   __restrict__ ref,
                       int t,
                       float* __restrict__ xT)
{
  __shared__ float s_score[256];
  __shared__ float s_red[256];
  __shared__ float s_ctx[FDIM];
  __shared__ float s_mx, s_sum;

  const int b   = blockIdx.x;
  const int tid = threadIdx.x;

  if (tid < LSEQ) {
    const float4* ep = (const float4*)(enc_proj + ((size_t)b * LSEQ + tid) * ADIM);
    const float4* hw = (const float4*)(hWa + (size_t)b * ADIM);
    const float4* va = (const float4*)v_att;
    float acc = 0.0f;
    for (int k4 = 0; k4 < ADIM / 4; ++k4) {
      float4 e = ep[k4], h = hw[k4], v = va[k4];
      acc += v.x * tanhf(e.x + h.x) + v.y * tanhf(e.y + h.y)
           + v.z * tanhf(e.z + h.z) + v.w * tanhf(e.w + h.w);
    }
    float m = mask[(size_t)b * LSEQ + tid];
    s_score[tid] = (m > 0.0f) ? acc : -1e9f;
  } else {
    s_score[tid] = -1e30f;  // exp -> 0
  }
  __syncthreads();

  s_red[tid] = s_score[tid];
  __syncthreads();
  for (int s = 128; s > 0; s >>= 1) {
    if (tid < s) s_red[tid] = fmaxf(s_red[tid], s_red[tid + s]);
    __syncthreads();
  }
  if (tid == 0) s_mx = s_red[0];
  __syncthreads();

  float e = expf(s_score[tid] - s_mx);
  s_score[tid] = e;
  s_red[tid] = e;
  __syncthreads();
  for (int s = 128; s > 0; s >>= 1) {
    if (tid < s) s_red[tid] += s_red[tid + s];
    __syncthreads();
  }
  if (tid == 0) s_sum = s_red[0];
  __syncthreads();

  const float inv = 1.0f / s_sum;
  for (int f = tid; f < FDIM; f += 256) {
    float acc = 0.0f;
    const float* ebase = enc + (size_t)b * LSEQ * FDIM + f;
    for (int l = 0; l < LSEQ; ++l)
      acc += s_score[l] * ebase[(size_t)l * FDIM];
    s_ctx[f] = acc * inv;
  }
  __syncthreads();

  const int word = ref[b * TSZ + t];
  for (int k = tid; k < DDIM; k += 256) {
    float v = (k < EDIM) ? emb[(size_t)word * EDIM + k] : s_ctx[k - EDIM];
    xT[(size_t)k * BSZ + b] = v;
  }
}

// GRU gate math. giT/ghT are [3H x B] (gate-major from the GEMMs).
__global__ void k_gru_gate(const float* __restrict__ giT,
                           const float* __restrict__ ghT,
                           const float* __restrict__ h,
                           float* __restrict__ h_new,
                           float* __restrict__ hT_new)
{
  int i = blockIdx.x * blockDim.x + threadIdx.x;  // BSZ*HDIM
  int b = i >> 9;
  int u = i & (HDIM - 1);
  float ir  = giT[(size_t)u * BSZ + b];
  float iz  = giT[(size_t)(HDIM + u) * BSZ + b];
  float in_ = giT[(size_t)(2 * HDIM + u) * BSZ + b];
  float hr  = ghT[(size_t)u * BSZ + b];
  float hz  = ghT[(size_t)(HDIM + u) * BSZ + b];
  float hn  = ghT[(size_t)(2 * HDIM + u) * BSZ + b];
  float r = 1.0f / (1.0f + expf(-(ir + hr)));
  float z = 1.0f / (1.0f + expf(-(iz + hz)));
  float n = tanhf(in_ + r * hn);
  float hv = h[(size_t)b * HDIM + u];
  float o = (1.0f - z) * n + z * hv;
  h_new[(size_t)b * HDIM + u] = o;
  hT_new[(size_t)u * BSZ + b] = o;
}

__global__ void k_transpose_h(const float* __restrict__ h, float* __restrict__ hT)
{
  int i = blockIdx.x * blockDim.x + threadIdx.x;  // BSZ*HDIM
  int b = i >> 9;
  int u = i & (HDIM - 1);
  hT[(size_t)u * BSZ + b] = h[(size_t)b * HDIM + u];
}

// log-sum-exp per batch row over V (logits stored [V x B]).
__global__ void k_lse(const float* __restrict__ logitsT, float* __restrict__ lse)
{
  __shared__ float s_red[256];
  __shared__ float s_mx;
  const int b = blockIdx.x, tid = threadIdx.x;
  float mx = -1e30f;
  for (int v = tid; v < VSZ; v += 256) mx = fmaxf(mx, logitsT[(size_t)v * BSZ + b]);
  s_red[tid] = mx;
  __syncthreads();
  for (int s = 128; s > 0; s >>= 1) {
    if (tid < s) s_red[tid] = fmaxf(s_red[tid], s_red[tid + s]);
    __syncthreads();
  }
  if (tid == 0) s_mx = s_red[0];
  __syncthreads();
  float sum = 0.0f;
  for (int v = tid; v < VSZ; v += 256) sum += expf(logitsT[(size_t)v * BSZ + b] - s_mx);
  s_red[tid] = sum;
  __syncthreads();
  for (int s = 128; s > 0; s >>= 1) {
    if (tid < s) s_red[tid] += s_red[tid + s];
    __syncthreads();
  }
  if (tid == 0) lse[b] = s_mx + logf(s_red[0]);
}

// out[b, t, v] = logitsT[v][b] - lse[b]   (writes coalesced over v)
__global__ void k_writeout(const float* __restrict__ logitsT,
                           const float* __restrict__ lse,
                           float* __restrict__ out, int t)
{
  size_t i = (size_t)blockIdx.x * blockDim.x + threadIdx.x;
  if (i >= (size_t)BSZ * VSZ) return;
  int b = (int)(i / VSZ);
  int v = (int)(i % VSZ);
  out[((size_t)b * TSZ + t) * VSZ + v] = logitsT[(size_t)v * BSZ + b] - lse[b];
}

__global__ void k_zero_first(float* __restrict__ out)
{
  size_t i = (size_t)blockIdx.x * blockDim.x + threadIdx.x;
  if (i >= (size_t)BSZ * VSZ) return;
  int b = (int)(i / VSZ);
  int v = (int)(i % VSZ);
  out[(size_t)b * TSZ * VSZ + v] = 0.0f;
}

// ---------------------------------------------------------------------------
extern "C" void kernel_launch(void* const* d_in, const int* in_sizes, int n_in,
                              void* d_out, int out_size, void* d_ws, size_t ws_size,
                              hipStream_t stream)
{
  const float* enc     = (const float*)d_in[0];   // (B,L,F)
  const float* h_init  = (const float*)d_in[1];   // (1,B,H)
  const float* mask    = (const float*)d_in[2];   // (B,L,1)
  // d_in[3] temp_seg unused by reference forward
  const float* emb     = (const float*)d_in[4];   // (V,E)
  const float* W_ih    = (const float*)d_in[5];   // (3H,D)
  const float* W_hh    = (const float*)d_in[6];   // (3H,H)
  const float* b_ih    = (const float*)d_in[7];
  const float* b_hh    = (const float*)d_in[8];
  const float* Wa_enc  = (const float*)d_in[9];   // (F,A)
  const float* Wa_h    = (const float*)d_in[10];  // (H,A)
  const float* v_att   = (const float*)d_in[11];  // (A,)
  const float* W_out   = (const float*)d_in[12];  // (V,H)
  const float* b_out   = (const float*)d_in[13];  // (V,)
  const int*   ref     = (const int*)d_in[14];    // (B,T)
  float* out = (float*)d_out;

  char* p = (char*)d_ws;
  float* enc_proj = (float*)p; p += (size_t)BSZ * LSEQ * ADIM * 4;  // 26.2 MB
  float* hWa      = (float*)p; p += (size_t)BSZ * ADIM * 4;
  float* xT       = (float*)p; p += (size_t)DDIM * BSZ * 4;
  float* giT      = (float*)p; p += (size_t)G3 * BSZ * 4;
  float* ghT      = (float*)p; p += (size_t)G3 * BSZ * 4;
  float* hA       = (float*)p; p += (size_t)BSZ * HDIM * 4;
  float* hB       = (float*)p; p += (size_t)BSZ * HDIM * 4;
  float* hT       = (float*)p; p += (size_t)HDIM * BSZ * 4;
  float* logitsT  = (float*)p; p += (size_t)VSZ * BSZ * 4;          // 12.9 MB
  float* lse      = (float*)p; p += (size_t)BSZ * 4;

  const int TPB = 256, WPB = TPB / 32;
  const int MT = 4;
  auto blocks_for = [&](int M, int N) {
    int mgroups = (M + 16 * MT - 1) / (16 * MT);
    int waves = mgroups * (N / 16);
    return (waves + WPB - 1) / WPB;
  };

  // d_out[:,0,:] = 0  (BOS slot)
  {
    int blocks = (int)(((size_t)BSZ * VSZ + TPB - 1) / TPB);
    k_zero_first<<<blocks, TPB, 0, stream>>>(out);
  }

  // enc_proj = encoder_output @ Wa_enc : M=12800, N=512, K=512
  k_gemm<MT, FDIM, ADIM, ADIM, false>
      <<<blocks_for(BSZ * LSEQ, ADIM), TPB, 0, stream>>>(
          enc, Wa_enc, enc_proj, nullptr, BSZ * LSEQ, ADIM, FDIM);

  // hT = h_init^T (for first-step gh GEMM)
  k_transpose_h<<<(BSZ * HDIM) / TPB, TPB, 0, stream>>>(h_init, hT);

  const float* h_cur = h_init;
  float* hbuf[2] = {hA, hB};

  for (int t = 0; t < TSZ - 1; ++t) {
    // hWa = h @ Wa_h : M=64, N=512, K=512
    k_gemm<MT, HDIM, ADIM, ADIM, false>
        <<<blocks_for(BSZ, ADIM), TPB, 0, stream>>>(
            h_cur, Wa_h, hWa, nullptr, BSZ, ADIM, HDIM);

    // attention + softmax + ctx + concat -> xT (D x B)
    k_attn<<<BSZ, TPB, 0, stream>>>(enc_proj, hWa, v_att, mask, enc, emb, ref, t, xT);

    // giT = W_ih @ xT + b_ih : M=1536, N=64, K=1024
    k_gemm<MT, DDIM, BSZ, BSZ, true>
        <<<blocks_for(G3, BSZ), TPB, 0, stream>>>(
            W_ih, xT, giT, b_ih, G3, BSZ, DDIM);

    // ghT = W_hh @ hT + b_hh : M=1536, N=64, K=512
    k_gemm<MT, HDIM, BSZ, BSZ, true>
        <<<blocks_for(G3, BSZ), TPB, 0, stream>>>(
            W_hh, hT, ghT, b_hh, G3, BSZ, HDIM);

    // gates -> h_next (B x H) and hT (H x B)
    float* h_next = hbuf[t & 1];
    k_gru_gate<<<(BSZ * HDIM) / TPB, TPB, 0, stream>>>(giT, ghT, h_cur, h_next, hT);

    // logitsT = W_out @ hT + b_out : M=V=50257, N=64, K=512
    k_gemm<MT, HDIM, BSZ, BSZ, true>
        <<<blocks_for(VSZ, BSZ), TPB, 0, stream>>>(
            W_out, hT, logitsT, b_out, VSZ, BSZ, HDIM);

    // log-softmax
    k_lse<<<BSZ, TPB, 0, stream>>>(logitsT, lse);
    {
      int blocks = (int)(((size_t)BSZ * VSZ + TPB - 1) / TPB);
      k_writeout<<<blocks, TPB, 0, stream>>>(logitsT, lse, out, t + 1);
    }

    h_cur = h_next;
  }
}